// STULayer_48550310314370
// MI455X (gfx1250) — compile-verified
//
#include <hip/hip_runtime.h>
#include <math.h>

#define B_  8
#define D_  256
#define L_  2048
#define K_  24
#define R_  1025
#define H_  512

typedef float v2f  __attribute__((ext_vector_type(2)));
typedef float v8f  __attribute__((ext_vector_type(8)));
typedef unsigned int u32x4 __attribute__((ext_vector_type(4)));
typedef int   i32x4 __attribute__((ext_vector_type(4)));
typedef int   i32x8 __attribute__((ext_vector_type(8)));

#if __has_builtin(__builtin_amdgcn_tensor_load_to_lds)
#define HAVE_TDM 1
#else
#define HAVE_TDM 0
#endif

// ---------------------------------------------------------------------------
// TDM helpers: build D# groups and issue tensor_load_to_lds.
//   group0: count=1 | lds_addr[63:32] | global_addr[120:64] | type=2[127:126]
//   group1: data_size/pad | tensor_dim0/1 | tile_dim0/1/2 | strides
// ---------------------------------------------------------------------------
__device__ __forceinline__ unsigned stu_lds_offset(const void* p) {
    // generic LDS address = {SHARED_BASE, 32'offset}; low 32 bits = LDS offset
    return (unsigned)(unsigned long long)p;
}

__device__ __forceinline__ void stu_tdm_wait0() {
#if __has_builtin(__builtin_amdgcn_s_wait_tensorcnt)
    __builtin_amdgcn_s_wait_tensorcnt(0);
#else
    asm volatile("s_wait_tensorcnt 0" ::: "memory");
#endif
}

#if HAVE_TDM
__device__ __forceinline__ void stu_tdm_load_2d(
    unsigned ldsAddr, const void* gptr,
    unsigned tensorDim0, unsigned tensorDim1,
    unsigned tileDim0, unsigned tileDim1,
    unsigned long long stride0Elems,
    unsigned padded /* 0: none, 1: +2 dwords every 16 dwords */) {

    unsigned long long ga = (unsigned long long)gptr;

    u32x4 g0;
    g0.x = 1u;                                   // count=1 (valid descriptor)
    g0.y = ldsAddr;                              // lds_addr [63:32]
    g0.z = (unsigned)(ga & 0xffffffffu);         // global_addr [95:64]
    g0.w = (unsigned)((ga >> 32) & 0x1ffffffu)   // global_addr [120:96]
         | 0x80000000u;                          // type=2 [127:126]

    unsigned w0 = (2u << 16);                    // data_size = 4 bytes
    if (padded) w0 |= (1u << 20)                 // pad_enable
               |  (3u << 22)                     // pad_interval: 16 DWORDs
               |  (1u << 25);                    // pad_amount:   2 DWORDs
    i32x8 g1;
    g1.s0 = (int)w0;
    g1.s1 = (int)((tensorDim0 & 0xffffu) << 16);             // dim0[15:0]
    g1.s2 = (int)((tensorDim0 >> 16) |
                  ((tensorDim1 & 0xffffu) << 16));           // dim0[31:16]|dim1[15:0]
    g1.s3 = (int)((tensorDim1 >> 16) |
                  ((tileDim0 & 0xffffu) << 16));             // dim1[31:16]|tile0
    g1.s4 = (int)(tileDim1 & 0xffffu);                       // tile1 (tile2=0)
    g1.s5 = (int)(unsigned)(stride0Elems & 0xffffffffull);   // stride0[31:0]
    g1.s6 = (int)(unsigned)((stride0Elems >> 32) & 0xffffu); // stride0[47:32]
    g1.s7 = 0;

    i32x4 z4 = {0, 0, 0, 0};
#if __clang_major__ >= 23
    i32x8 z8 = {0, 0, 0, 0, 0, 0, 0, 0};
    __builtin_amdgcn_tensor_load_to_lds(g0, g1, z4, z4, z8, 0);
#else
    __builtin_amdgcn_tensor_load_to_lds(g0, g1, z4, z4, 0);
#endif
}
#endif

// ---------------------------------------------------------------------------
// Kernel 1: LayerNorm over channel dim D. One thread per (b,l) column.
// ---------------------------------------------------------------------------
__global__ void stu_ln_kernel(const float* __restrict__ x,
                              const float* __restrict__ gamma,
                              const float* __restrict__ beta,
                              float* __restrict__ z) {
    int t = blockIdx.x * blockDim.x + threadIdx.x;   // 0 .. B*L-1
    int b = t / L_;
    int l = t - b * L_;
    const float* xp = x + (size_t)b * D_ * L_ + l;
    float s = 0.f, s2 = 0.f;
    for (int d = 0; d < D_; ++d) {
        float v = xp[(size_t)d * L_];
        s += v; s2 += v * v;
    }
    float mu  = s  * (1.0f / D_);
    float var = s2 * (1.0f / D_) - mu * mu;
    float rs  = rsqrtf(var + 1e-5f);
    float* zp = z + (size_t)b * D_ * L_ + l;
    for (int d = 0; d < D_; ++d) {
        zp[(size_t)d * L_] = (xp[(size_t)d * L_] - mu) * rs * gamma[d] + beta[d];
    }
}

// ---------------------------------------------------------------------------
// Radix-2 Stockham FFT, 2048 points, in LDS. 256 threads x 4 butterflies,
// 11 stages, natural-order output.
// ---------------------------------------------------------------------------
template <bool INVERSE>
__device__ inline float2* stu_fft2048(float2* bufA, float2* bufB, int t) {
    float2* cur = bufA;
    float2* nxt = bufB;
    int n = 2048, s = 1;
    while (n >= 2) {
        int   m      = n >> 1;
        float theta0 = 6.28318530717958647692f / (float)n;
        for (int i = 0; i < 4; ++i) {
            int idx = t + 256 * i;          // 0..1023  (m*s == 1024 each stage)
            int p   = idx / s;
            int q   = idx - p * s;
            float2 a  = cur[q + s * p];
            float2 bv = cur[q + s * (p + m)];
            float si, c;
            __sincosf((float)p * theta0, &si, &c);
            float2 sum = make_float2(a.x + bv.x, a.y + bv.y);
            float2 dif = make_float2(a.x - bv.x, a.y - bv.y);
            float2 tw;
            if (INVERSE) {
                tw = make_float2(dif.x * c - dif.y * si, dif.y * c + dif.x * si);
            } else {
                tw = make_float2(dif.x * c + dif.y * si, dif.y * c - dif.x * si);
            }
            nxt[q + s * (2 * p)]     = sum;
            nxt[q + s * (2 * p + 1)] = tw;
        }
        __syncthreads();
        float2* tmp = cur; cur = nxt; nxt = tmp;
        n >>= 1; s <<= 1;
    }
    return cur;
}

// ---------------------------------------------------------------------------
// Kernel 2: forward rfft per (b,d) row. Stores transposed spectrum d-major:
// xfT[r][d*8 + b] (float2) -> per-r tiles are TDM-friendly contiguous rows
// and bank-conflict-free in LDS without padding.
// ---------------------------------------------------------------------------
__global__ void stu_fft_fwd_kernel(const float* __restrict__ z,
                                   float2* __restrict__ xfT) {
    __shared__ float2 bufA[2048];
    __shared__ float2 bufB[2048];
    int row = blockIdx.x;             // b*256 + d
    int t   = threadIdx.x;
    const float* zp = z + (size_t)row * L_;
    for (int i = 0; i < 8; ++i) {
        int idx = t + 256 * i;
        bufA[idx] = make_float2(zp[idx], 0.0f);
    }
    __syncthreads();
    float2* res = stu_fft2048<false>(bufA, bufB, t);
    int b = row >> 8, d = row & 255;
    for (int i = 0; i < 5; ++i) {
        int r = t + 256 * i;
        if (r < R_) xfT[(size_t)r * 2048 + d * 8 + b] = res[r];
    }
}

// ---------------------------------------------------------------------------
// Kernel 3: spectral mixing with f32 WMMA.
//   Sf[b,h,r] = sum_k Theta_k @ (Phi_c[k,r] * Xf[b,:,r])
// Xf tiles staged via Tensor Data Mover (2D descriptor, OOB rows -> zeros).
// ---------------------------------------------------------------------------
#define NR 3

__global__ void stu_spectral_kernel(const float2* __restrict__ xfT,
                                    const float*  __restrict__ theta,
                                    const float*  __restrict__ phiRe,
                                    const float*  __restrict__ phiIm,
                                    float* __restrict__ sfC) {
    __shared__ float2 xf[NR * 2048];          // [rr][d*8+b], 48 KB contiguous
    int t  = threadIdx.x;
    int r0 = blockIdx.x * NR;

#if HAVE_TDM
    if (t < 32) {                             // wave 0 drives the TDM
        unsigned rows = (unsigned)(R_ - r0);  // ragged tail: OOB rows -> 0
        stu_tdm_load_2d(stu_lds_offset(&xf[0]),
                        (const void*)(xfT + (size_t)r0 * 2048),
                        /*tensorDim0=*/4096, /*tensorDim1=*/rows,
                        /*tileDim0=*/4096,   /*tileDim1=*/NR,
                        /*stride0=*/4096ull, /*padded=*/0);
        stu_tdm_wait0();
    }
#else
    for (int rr = 0; rr < NR; ++rr) {
        int r = r0 + rr;
        for (int i = 0; i < 8; ++i) {
            int idx = t + 256 * i;
            xf[rr * 2048 + idx] = (r < R_) ? xfT[(size_t)r * 2048 + idx]
                                           : make_float2(0.f, 0.f);
        }
    }
#endif
    __syncthreads();

    int wave = t >> 5;
    int lane = t & 31;
    int n16  = lane & 15;     // column within 16-wide tile
    int hi   = lane >> 4;     // half-wave select
    int bb   = n16 >> 1;      // batch index  (col = 2*b + e)
    int ee   = n16 & 1;       // 0 = real col, 1 = imag col
    int m0   = wave * 32;     // this wave's h base (2 M-tiles)

    v8f acc[2][NR];
    const v8f vzero = {0.f,0.f,0.f,0.f,0.f,0.f,0.f,0.f};
    for (int mt = 0; mt < 2; ++mt)
        for (int rr = 0; rr < NR; ++rr) acc[mt][rr] = vzero;

    for (int k = 0; k < K_; ++k) {
        float pr[NR], pim[NR];
        for (int rr = 0; rr < NR; ++rr) {
            int r = r0 + rr;
            bool ok = (r < R_);
            pr[rr]  = ok ? phiRe[k * R_ + r] : 0.f;
            pim[rr] = ok ? phiIm[k * R_ + r] : 0.f;
        }
        const float2* th2 = (const float2*)(theta + (size_t)k * D_ * D_);
        for (int d0 = 0; d0 < D_; d0 += 4) {
            v2f bfr[NR];
            for (int rr = 0; rr < NR; ++rr) {
                float2 x0 = xf[rr * 2048 + (d0 + 2 * hi) * 8 + bb];
                float2 x1 = xf[rr * 2048 + (d0 + 2 * hi + 1) * 8 + bb];
                // Phi_c = pr - i*pim ; col e=0: Re(Phi_c*X), e=1: Im(Phi_c*X)
                float b0 = ee ? (pr[rr] * x0.y - pim[rr] * x0.x)
                              : (pr[rr] * x0.x + pim[rr] * x0.y);
                float b1 = ee ? (pr[rr] * x1.y - pim[rr] * x1.x)
                              : (pr[rr] * x1.x + pim[rr] * x1.y);
                bfr[rr].x = b0; bfr[rr].y = b1;
            }
            for (int mt = 0; mt < 2; ++mt) {
                int h = m0 + mt * 16 + n16;   // A row for this lane
                float2 av = th2[(size_t)h * (D_ / 2) + ((d0 >> 1) + hi)];
                v2f afr; afr.x = av.x; afr.y = av.y;
                for (int rr = 0; rr < NR; ++rr) {
                    acc[mt][rr] = __builtin_amdgcn_wmma_f32_16x16x4_f32(
                        false, afr, false, bfr[rr], (short)0, acc[mt][rr],
                        false, false);
                }
            }
        }
    }

    for (int mt = 0; mt < 2; ++mt) {
        for (int rr = 0; rr < NR; ++rr) {
            int r = r0 + rr;
            if (r >= R_) continue;
            for (int i = 0; i < 8; ++i) {
                int h = m0 + mt * 16 + hi * 8 + i;
                sfC[(((size_t)ee * 8 + bb) * 256 + h) * R_ + r] = acc[mt][rr][i];
            }
        }
    }
}

// ---------------------------------------------------------------------------
// Kernel 4: irfft per (b,h) row. Hermitian-extend Sf, inverse FFT, Re()/N.
// ---------------------------------------------------------------------------
__global__ void stu_fft_inv_kernel(const float* __restrict__ sfC,
                                   float* __restrict__ S) {
    __shared__ float2 bufA[2048];
    __shared__ float2 bufB[2048];
    int row = blockIdx.x;              // b*256 + h
    int t   = threadIdx.x;
    const float* re = sfC + (size_t)row * R_;
    const float* im = sfC + (size_t)(row + 2048) * R_;   // e=1 plane
    for (int i = 0; i < 5; ++i) {
        int r = t + 256 * i;
        if (r < R_) {
            float xr = re[r], xi = im[r];
            bufA[r] = make_float2(xr, xi);
            if (r >= 1 && r <= 1023) bufA[2048 - r] = make_float2(xr, -xi);
        }
    }
    __syncthreads();
    float2* res = stu_fft2048<true>(bufA, bufB, t);
    float* sp = S + (size_t)row * L_;
    for (int i = 0; i < 8; ++i) {
        int l = t + 256 * i;
        sp[l] = res[l].x * (1.0f / 2048.0f);
    }
}

// ---------------------------------------------------------------------------
// Kernel 5: fused 1x1-conv MLP with f32 WMMA + exact GELU + residual.
// S tile staged via TDM with hardware padding (16 dwords + 2 pad = stride 18).
// ---------------------------------------------------------------------------
#define SP 18   // LDS row stride (floats)

__global__ void stu_mlp_kernel(const float* __restrict__ S,
                               const float* __restrict__ x,
                               const float* __restrict__ W1,
                               const float* __restrict__ b1v,
                               const float* __restrict__ W2,
                               const float* __restrict__ b2v,
                               float* __restrict__ out) {
    __shared__ float sTile[D_ * SP];
    __shared__ float hTile[H_ * SP];
    __shared__ float b1s[H_];
    __shared__ float b2s[D_];

    int t  = threadIdx.x;
    int wg = blockIdx.x;               // 1024 workgroups
    int b  = wg >> 7;                  // 128 col-tiles per batch
    int l0 = (wg & 127) << 4;

    b1s[t]       = b1v[t];
    b1s[t + 256] = b1v[t + 256];
    b2s[t]       = b2v[t];

#if HAVE_TDM
    if (t < 32) {                      // wave 0 drives the TDM
        stu_tdm_load_2d(stu_lds_offset(&sTile[0]),
                        (const void*)(S + (size_t)b * D_ * L_ + l0),
                        /*tensorDim0=*/L_,  /*tensorDim1=*/D_,
                        /*tileDim0=*/16,    /*tileDim1=*/D_,
                        /*stride0=*/(unsigned long long)L_, /*padded=*/1);
        stu_tdm_wait0();
    }
#else
    {
        const float* sp = S + ((size_t)b * D_ + t) * L_ + l0;  // t == d row
        for (int c = 0; c < 16; ++c) sTile[t * SP + c] = sp[c];
    }
#endif
    __syncthreads();

    int wave = t >> 5, lane = t & 31;
    int n16 = lane & 15, hi = lane >> 4;

    // warm L2 for this wave's W2 panel while GEMM1 runs
    __builtin_prefetch(&W2[(size_t)(wave * 2) * 16 * H_], 0, 1);
    __builtin_prefetch(&W2[(size_t)(wave * 2 + 1) * 16 * H_], 0, 1);

    const v8f vzero = {0.f,0.f,0.f,0.f,0.f,0.f,0.f,0.f};
    const float2* W1v = (const float2*)W1;
    const float2* W2v = (const float2*)W2;

    // GEMM1: h = W1 @ Stile ; each wave owns 4 M-tiles of 32 total.
    for (int mi = 0; mi < 4; ++mi) {
        int h0 = (wave * 4 + mi) * 16;
        v8f acc = vzero;
        for (int d0 = 0; d0 < D_; d0 += 4) {
            float2 av = W1v[(size_t)(h0 + n16) * (D_ / 2) + ((d0 >> 1) + hi)];
            v2f afr; afr.x = av.x; afr.y = av.y;
            v2f bfr;
            bfr.x = sTile[(d0 + 2 * hi) * SP + n16];
            bfr.y = sTile[(d0 + 2 * hi + 1) * SP + n16];
            acc = __builtin_amdgcn_wmma_f32_16x16x4_f32(
                false, afr, false, bfr, (short)0, acc, false, false);
        }
        for (int i = 0; i < 8; ++i) {
            int h = h0 + hi * 8 + i;
            float v = acc[i] + b1s[h];
            v = 0.5f * v * (1.0f + erff(v * 0.70710678118654752440f));
            hTile[h * SP + n16] = v;
        }
    }
    __syncthreads();

    // GEMM2: out = W2 @ h ; each wave owns 2 M-tiles of 16 total.
    for (int mi = 0; mi < 2; ++mi) {
        int d0r = (wave * 2 + mi) * 16;
        v8f acc = vzero;
        for (int k0 = 0; k0 < H_; k0 += 4) {
            float2 av = W2v[(size_t)(d0r + n16) * (H_ / 2) + ((k0 >> 1) + hi)];
            v2f afr; afr.x = av.x; afr.y = av.y;
            v2f bfr;
            bfr.x = hTile[(k0 + 2 * hi) * SP + n16];
            bfr.y = hTile[(k0 + 2 * hi + 1) * SP + n16];
            acc = __builtin_amdgcn_wmma_f32_16x16x4_f32(
                false, afr, false, bfr, (short)0, acc, false, false);
        }
        for (int i = 0; i < 8; ++i) {
            int d = d0r + hi * 8 + i;
            size_t off = ((size_t)b * D_ + d) * L_ + l0 + n16;
            out[off] = x[off] + acc[i] + b2s[d];
        }
    }
}

// ---------------------------------------------------------------------------
// Host launcher
// ---------------------------------------------------------------------------
extern "C" void kernel_launch(void* const* d_in, const int* in_sizes, int n_in,
                              void* d_out, int out_size, void* d_ws, size_t ws_size,
                              hipStream_t stream) {
    const float* x     = (const float*)d_in[0];
    const float* phiRe = (const float*)d_in[1];
    const float* phiIm = (const float*)d_in[2];
    const float* theta = (const float*)d_in[3];
    const float* gamma = (const float*)d_in[4];
    const float* beta  = (const float*)d_in[5];
    const float* W1    = (const float*)d_in[6];
    const float* b1v   = (const float*)d_in[7];
    const float* W2    = (const float*)d_in[8];
    const float* b2v   = (const float*)d_in[9];
    float* out = (float*)d_out;

    char* ws = (char*)d_ws;
    const size_t zBytes   = (size_t)B_ * D_ * L_ * sizeof(float);        // 16 MB
    const size_t xfBytes  = (size_t)R_ * 2048 * sizeof(float2);          // 16 MB
    const size_t sfBytes  = (size_t)2 * B_ * D_ * R_ * sizeof(float);    // 16 MB
    float*  z   = (float*)(ws);
    float2* xfT = (float2*)(ws + zBytes);
    float*  sfC = (float*)(ws + zBytes + xfBytes);
    float*  S   = (float*)(ws + zBytes + xfBytes + sfBytes);

    stu_ln_kernel<<<(B_ * L_) / 256, 256, 0, stream>>>(x, gamma, beta, z);
    stu_fft_fwd_kernel<<<B_ * D_, 256, 0, stream>>>(z, xfT);
    stu_spectral_kernel<<<(R_ + NR - 1) / NR, 256, 0, stream>>>(
        xfT, theta, phiRe, phiIm, sfC);
    stu_fft_inv_kernel<<<B_ * D_, 256, 0, stream>>>(sfC, S);
    stu_mlp_kernel<<<(B_ * L_) / 16, 256, 0, stream>>>(
        S, x, W1, b1v, W2, b2v, out);
}